// LinOSSLayer_25821343383627
// MI455X (gfx1250) — compile-verified
//
#include <hip/hip_runtime.h>
#include <math.h>

// LinOSS (IM discretization) forward on gfx1250.
// Bt=16, L=4096, H=128, P=256.  GEMMs via v_wmma_f32_16x16x4_f32 (exact f32).
// Scan over L chunked (32 x 128) using time-invariant 2x2 M; carries combined
// with M^128 (7 squarings).  bu/states kept in one in-place buffer.

typedef __attribute__((ext_vector_type(2))) float v2f;
typedef __attribute__((ext_vector_type(8))) float v8f;

#define BT    16
#define LSEQ  4096
#define HD    128
#define PD    256
#define NROWS (BT * LSEQ)       // 65536
#define NC    32
#define CHUNK (LSEQ / NC)       // 128

// ---- workspace layout (float offsets) ----
#define O_COEF   ((size_t)0)                          // 16 floats per p
#define O_BPK    ((size_t)4096)                       // B packed [part][h/2][p][2]
#define O_CPK    (O_BPK + (size_t)2 * HD * PD)        // C packed [part][p/2][h][2] (im negated)
#define O_BURE   (O_CPK + (size_t)2 * HD * PD)        // bu_re / states_re [row][p]
#define O_BUIM   (O_BURE + (size_t)NROWS * PD)        // bu_im / states_im [row][p]
#define O_CARRY  (O_BUIM + (size_t)NROWS * PD)        // (b,c,part,p) x {x1,x2}
#define O_PREFIX (O_CARRY + (size_t)BT * NC * 2 * PD * 2)
// total ~ 34.74M floats = ~133 MiB workspace

// ---- optional gfx1250 async global->LDS path (probe at compile time) ----
#if defined(__has_builtin)
# if __has_builtin(__builtin_amdgcn_global_load_async_to_lds_b128) && \
     __has_builtin(__builtin_amdgcn_s_wait_asynccnt)
#  define USE_ASYNC_LDS 1
# endif
#endif
#ifndef USE_ASYNC_LDS
# define USE_ASYNC_LDS 0
#endif

#if USE_ASYNC_LDS
// Builtin signature (from hipcc diagnostic): params are pointers to
// int __attribute__((vector_size(16))) in global / LDS address spaces.
typedef int v4i_ __attribute__((vector_size(16)));
typedef __attribute__((address_space(1))) v4i_ gv4i;
typedef __attribute__((address_space(3))) v4i_ sv4i;
__device__ __forceinline__ void async_b128(const float* g, float* l) {
    __builtin_amdgcn_global_load_async_to_lds_b128((gv4i*)g, (sv4i*)l, 0, 0);
}
__device__ __forceinline__ void async_wait0() {
    __builtin_amdgcn_s_wait_asynccnt(0);
}
#endif

// ---------------------------------------------------------------- coefficients
__global__ void k_coef(const float* __restrict__ A_diag,
                       const float* __restrict__ steps,
                       float* __restrict__ ws) {
    int p = threadIdx.x;                    // 256 threads, 1 block
    float a  = fmaxf(A_diag[p], 0.0f);
    float st = 1.0f / (1.0f + __expf(-steps[p]));
    float s2a   = st * st * a;
    float schur = 1.0f / (1.0f + s2a);
    float m11 = 1.0f - s2a * schur;
    float m12 = -st * a * schur;
    float m21 = st * schur;
    float m22 = schur;
    float q11 = m11, q12 = m12, q21 = m21, q22 = m22;   // M^128: 7 squarings
#pragma unroll
    for (int i = 0; i < 7; ++i) {
        float n11 = q11 * q11 + q12 * q21;
        float n12 = q11 * q12 + q12 * q22;
        float n21 = q21 * q11 + q22 * q21;
        float n22 = q21 * q12 + q22 * q22;
        q11 = n11; q12 = n12; q21 = n21; q22 = n22;
    }
    float* c = ws + O_COEF + (size_t)p * 16;
    c[0] = m11 * st;  // g1
    c[1] = m21 * st;  // g2
    c[2] = m11; c[3] = m12; c[4] = m21; c[5] = m22;
    c[6] = q11; c[7] = q12; c[8] = q21; c[9] = q22;
}

// --------------------------------- pack B, C into k-pair-interleaved layouts
__global__ void k_pack(const float* __restrict__ B,   // (P,H,2)
                       const float* __restrict__ C,   // (H,P,2)
                       float* __restrict__ ws) {
    int idx = blockIdx.x * 256 + threadIdx.x;         // 0 .. P*H-1
    int p = idx >> 7;          // /HD
    int h = idx & (HD - 1);
    float bre = B[((size_t)p * HD + h) * 2 + 0];
    float bim = B[((size_t)p * HD + h) * 2 + 1];
    float cre = C[((size_t)h * PD + p) * 2 + 0];
    float cim = C[((size_t)h * PD + p) * 2 + 1];
    size_t bofs = ((size_t)(h >> 1) * PD + p) * 2 + (h & 1);
    ws[O_BPK + bofs]                     = bre;
    ws[O_BPK + (size_t)HD * PD + bofs]   = bim;
    size_t cofs = ((size_t)(p >> 1) * HD + h) * 2 + (p & 1);
    ws[O_CPK + cofs]                     = cre;
    ws[O_CPK + (size_t)HD * PD + cofs]   = -cim;
}

// ------------------------------------------------ GEMM: bu = inputs @ B^T (c64)
__global__ void __launch_bounds__(256) k_gemm_bu(const float* __restrict__ u,
                                                 float* __restrict__ ws) {
    __shared__ float atile[16 * HD];                  // 8 KB
    const int tid = threadIdx.x;
    const int rowBase = blockIdx.x * 16;
    {
        const float4* src4 = (const float4*)(u + (size_t)rowBase * HD);
        float4* dst4 = (float4*)atile;
#if USE_ASYNC_LDS
        async_b128((const float*)(src4 + tid),       (float*)(dst4 + tid));
        async_b128((const float*)(src4 + tid + 256), (float*)(dst4 + tid + 256));
        async_wait0();
#else
        dst4[tid]       = src4[tid];
        dst4[tid + 256] = src4[tid + 256];
#endif
    }
    __syncthreads();

    const int wave = tid >> 5;
    const int lane = tid & 31;
    const int half = lane >> 4;     // 0: K lo pair, 1: K hi pair
    const int nn   = lane & 15;     // N (B/D lane index)
    const int mm   = lane & 15;     // M (A lane index)

    v8f zero = {};
    v8f acc[4];
#pragma unroll
    for (int i = 0; i < 4; ++i) acc[i] = zero;

#pragma unroll 8
    for (int ks = 0; ks < 32; ++ks) {
        const int kk = ks * 4 + 2 * half;      // even
        const int kp = kk >> 1;                // k-pair index
        v2f a;
        a.x = atile[mm * HD + kk];
        a.y = atile[mm * HD + kk + 1];
#pragma unroll
        for (int i = 0; i < 4; ++i) {
            const int t = wave * 4 + i;        // 0..31 col tiles
            const int pcol = ((t & 15) << 4) + nn;
            const float* Bp = ws + O_BPK + (size_t)(t >> 4) * (HD * PD);
            v2f b = *(const v2f*)(Bp + ((size_t)kp * PD + pcol) * 2);
            acc[i] = __builtin_amdgcn_wmma_f32_16x16x4_f32(
                false, a, false, b, (short)0, acc[i], false, false);
        }
    }
#pragma unroll
    for (int i = 0; i < 4; ++i) {
        const int t = wave * 4 + i;
        const int pcol = ((t & 15) << 4) + nn;
        float* dst = ws + O_BURE + (size_t)(t >> 4) * ((size_t)NROWS * PD);
#pragma unroll
        for (int r = 0; r < 8; ++r) {
            const int row = rowBase + r + 8 * half;
            dst[(size_t)row * PD + pcol] = acc[i][r];
        }
    }
}

// --------------------------------------------------------- scan pass A: chunks
__global__ void k_scanA(float* __restrict__ ws) {
    const int p = threadIdx.x;
    const int c = blockIdx.x, b = blockIdx.y, part = blockIdx.z;
    const float* cf = ws + O_COEF + (size_t)p * 16;
    const float g1 = cf[0], g2 = cf[1], m11 = cf[2], m12 = cf[3],
                m21 = cf[4], m22 = cf[5];
    const float* bu = ws + O_BURE + (size_t)part * ((size_t)NROWS * PD);
    size_t base = ((size_t)(b * LSEQ + c * CHUNK)) * PD + p;
    float x1 = 0.0f, x2 = 0.0f;
#pragma unroll 4
    for (int l = 0; l < CHUNK; ++l) {
        __builtin_prefetch(bu + base + (size_t)(l + 24) * PD, 0, 1);
        float v = bu[base + (size_t)l * PD];
        float n1 = fmaf(m11, x1, fmaf(m12, x2, g1 * v));
        float n2 = fmaf(m21, x1, fmaf(m22, x2, g2 * v));
        x1 = n1; x2 = n2;
    }
    float* cr = ws + O_CARRY +
                ((((size_t)b * NC + c) * 2 + part) * PD + p) * 2;
    cr[0] = x1; cr[1] = x2;
}

// ------------------------------------------- scan pass B: combine chunk carries
__global__ void k_scanB(float* __restrict__ ws) {
    const int p = threadIdx.x;
    const int b = blockIdx.x, part = blockIdx.y;
    const float* cf = ws + O_COEF + (size_t)p * 16;
    const float q11 = cf[6], q12 = cf[7], q21 = cf[8], q22 = cf[9];
    float s1 = 0.0f, s2 = 0.0f;
#pragma unroll
    for (int c = 0; c < NC; ++c) {
        size_t idx = ((((size_t)b * NC + c) * 2 + part) * PD + p) * 2;
        float* pf = ws + O_PREFIX + idx;
        pf[0] = s1; pf[1] = s2;                 // state entering chunk c
        const float* cr = ws + O_CARRY + idx;
        float t1 = fmaf(q11, s1, fmaf(q12, s2, cr[0]));
        float t2 = fmaf(q21, s1, fmaf(q22, s2, cr[1]));
        s1 = t1; s2 = t2;
    }
}

// ------------------------------- scan pass C: re-scan with prefix, in-place x2
__global__ void k_scanC(float* __restrict__ ws) {
    const int p = threadIdx.x;
    const int c = blockIdx.x, b = blockIdx.y, part = blockIdx.z;
    const float* cf = ws + O_COEF + (size_t)p * 16;
    const float g1 = cf[0], g2 = cf[1], m11 = cf[2], m12 = cf[3],
                m21 = cf[4], m22 = cf[5];
    const float* pf = ws + O_PREFIX +
                      ((((size_t)b * NC + c) * 2 + part) * PD + p) * 2;
    float x1 = pf[0], x2 = pf[1];
    float* bu = ws + O_BURE + (size_t)part * ((size_t)NROWS * PD);
    size_t base = ((size_t)(b * LSEQ + c * CHUNK)) * PD + p;
#pragma unroll 4
    for (int l = 0; l < CHUNK; ++l) {
        __builtin_prefetch(bu + base + (size_t)(l + 24) * PD, 0, 1);
        float v = bu[base + (size_t)l * PD];
        float n1 = fmaf(m11, x1, fmaf(m12, x2, g1 * v));
        float n2 = fmaf(m21, x1, fmaf(m22, x2, g2 * v));
        x1 = n1; x2 = n2;
        bu[base + (size_t)l * PD] = x2;         // states = x2, in place
    }
}

// ---------------- projection: out = Re(states @ C) + u*D, WMMA K=2x256
__global__ void __launch_bounds__(256) k_proj(const float* __restrict__ u,
                                              const float* __restrict__ D,
                                              const float* __restrict__ ws,
                                              float* __restrict__ out) {
    __shared__ float stile[2][16 * PD];   // 32 KB: [0]=re, [1]=im
    const int tid = threadIdx.x;
    const int rowBase = blockIdx.x * 16;
    {   // 16x256 tiles are fully contiguous in [row][p] layout
        const float4* r4 = (const float4*)(ws + O_BURE + (size_t)rowBase * PD);
        const float4* i4 = (const float4*)(ws + O_BUIM + (size_t)rowBase * PD);
        float4* dR = (float4*)stile[0];
        float4* dI = (float4*)stile[1];
#pragma unroll
        for (int j = 0; j < 4; ++j) {
#if USE_ASYNC_LDS
            async_b128((const float*)(r4 + tid + j * 256), (float*)(dR + tid + j * 256));
            async_b128((const float*)(i4 + tid + j * 256), (float*)(dI + tid + j * 256));
#else
            dR[tid + j * 256] = r4[tid + j * 256];
            dI[tid + j * 256] = i4[tid + j * 256];
#endif
        }
#if USE_ASYNC_LDS
        async_wait0();
#endif
    }
    __syncthreads();

    const int wave = tid >> 5;
    const int lane = tid & 31;
    const int half = lane >> 4;
    const int nn   = lane & 15;
    const int mm   = lane & 15;
    const int hcol = wave * 16 + nn;          // 8 waves cover H=128

    v8f acc = {};
#pragma unroll
    for (int part = 0; part < 2; ++part) {
        const float* Cp = ws + O_CPK + (size_t)part * (HD * PD); // im pre-negated
        const float* sT = stile[part];
#pragma unroll 8
        for (int ks = 0; ks < 64; ++ks) {     // K = 256
            const int kk = ks * 4 + 2 * half;
            const int kp = kk >> 1;
            v2f a;
            a.x = sT[mm * PD + kk];
            a.y = sT[mm * PD + kk + 1];
            v2f b = *(const v2f*)(Cp + ((size_t)kp * HD + hcol) * 2);
            acc = __builtin_amdgcn_wmma_f32_16x16x4_f32(
                false, a, false, b, (short)0, acc, false, false);
        }
    }
    const float dv = D[hcol];
#pragma unroll
    for (int r = 0; r < 8; ++r) {
        const int row = rowBase + r + 8 * half;
        const size_t o = (size_t)row * HD + hcol;
        out[o] = acc[r] + u[o] * dv;
    }
}

// ------------------------------------------------------------------- launcher
extern "C" void kernel_launch(void* const* d_in, const int* in_sizes, int n_in,
                              void* d_out, int out_size, void* d_ws, size_t ws_size,
                              hipStream_t stream) {
    (void)in_sizes; (void)n_in; (void)out_size; (void)ws_size;
    const float* u      = (const float*)d_in[0];
    const float* A_diag = (const float*)d_in[1];
    const float* B      = (const float*)d_in[2];
    const float* C      = (const float*)d_in[3];
    const float* D      = (const float*)d_in[4];
    const float* steps  = (const float*)d_in[5];
    float* ws  = (float*)d_ws;
    float* out = (float*)d_out;

    k_coef<<<1, 256, 0, stream>>>(A_diag, steps, ws);
    k_pack<<<(PD * HD) / 256, 256, 0, stream>>>(B, C, ws);
    k_gemm_bu<<<NROWS / 16, 256, 0, stream>>>(u, ws);
    dim3 gChunk(NC, BT, 2);
    k_scanA<<<gChunk, 256, 0, stream>>>(ws);
    dim3 gComb(BT, 2, 1);
    k_scanB<<<gComb, 256, 0, stream>>>(ws);
    k_scanC<<<gChunk, 256, 0, stream>>>(ws);
    k_proj<<<NROWS / 16, 256, 0, stream>>>(u, D, ws, out);
}